// ICoB_Attention_14267881357412
// MI455X (gfx1250) — compile-verified
//
#include <hip/hip_runtime.h>
#include <hip/hip_bf16.h>
#include <math.h>

typedef float v2f __attribute__((ext_vector_type(2)));
typedef float v8f __attribute__((ext_vector_type(8)));

constexpr int BATCH  = 4;
constexpr int DCH    = 64;        // per-branch channels
constexpr int OCH    = 192;       // 3*DCH (q,k,v)
constexpr int IMG_H  = 256;
constexpr int IMG_W  = 256;
constexpr size_t NPIX = (size_t)IMG_H * IMG_W;   // 65536
constexpr int HEADS  = 8;
constexpr float EPSN = 1e-12f;
constexpr int QSTR   = 260;       // padded LDS stride (bank-conflict-free fragments)

// ---- workspace layout (float offsets) ----
constexpr size_t OFF_QKV  = 0;                              // [zi=br*4+b][192][N]
constexpr size_t SZ_QKV   = 8ull * OCH * NPIX;
constexpr size_t OFF_V    = OFF_QKV + SZ_QKV;               // [zi][64][N]
constexpr size_t SZ_V     = 8ull * DCH * NPIX;
constexpr size_t OFF_GRAM = OFF_V + SZ_V;                   // [zi][hp=4][16][16]
constexpr size_t SZ_GRAM  = 8ull * 1024;
constexpr size_t OFF_NQ   = OFF_GRAM + SZ_GRAM;             // [zi][64]
constexpr size_t SZ_NQ    = 8ull * DCH;
constexpr size_t OFF_NK   = OFF_NQ + SZ_NQ;                 // [zi][64]
constexpr size_t SZ_NK    = 8ull * DCH;
constexpr size_t OFF_M    = OFF_NK + SZ_NK;                 // [zi][64][64]

// =============================================================
// K0: zero the global reduction accumulators (gram + nq + nk)
// =============================================================
__global__ void k0_zero(float* __restrict__ acc, int n) {
  int i = blockIdx.x * blockDim.x + threadIdx.x;
  if (i < n) acc[i] = 0.f;
}

// =============================================================
// K1: qkv 1x1 conv as GEMM: out[oc][n] = sum_ic W[oc][ic]*x[ic][n]
// WMMA f32 16x16x4.  One block: 96 output channels x 128 pixels.
// grid = (512 pixel-blocks, 16 = zi*2 + mhalf), block = 256
// =============================================================
__global__ __launch_bounds__(256) void k1_qkv_gemm(
    const float* __restrict__ x, const float* __restrict__ wq1,
    const float* __restrict__ wq2, float* __restrict__ qkv)
{
  __shared__ __align__(16) float sW[96 * 64];   // 24 KB
  __shared__ __align__(16) float sX[64 * 128];  // 32 KB
  const int pb  = blockIdx.x;
  const int zi2 = blockIdx.y;
  const int zi  = zi2 >> 1;          // br*4 + b
  const int mh  = zi2 & 1;           // which half of the 192 out channels
  const int br  = zi >> 2, b = zi & 3;
  const int t   = threadIdx.x;
  const size_t pbase = (size_t)pb * 128;

  const float* xsrc = x + ((size_t)b * 128 + (size_t)br * 64) * NPIX;
  const float* Wsrc = (br ? wq2 : wq1) + (size_t)mh * 96 * 64;
  float* out = qkv + (size_t)zi * OCH * NPIX + (size_t)mh * 96 * NPIX;

  for (int i = 0; i < 6; ++i) {                      // 96*64/4 = 1536 float4
    int i4 = t + i * 256;
    ((float4*)sW)[i4] = ((const float4*)Wsrc)[i4];
  }
  for (int i = 0; i < 8; ++i) {                      // 64*128/4 = 2048 float4
    int i4 = t + i * 256;
    int ic = i4 >> 5, c4 = i4 & 31;
    ((float4*)sX)[i4] = *(const float4*)(xsrc + (size_t)ic * NPIX + pbase + c4 * 4);
  }
  __syncthreads();

  const int wv = t >> 5, lane = t & 31;
  const int hf = lane >> 4, l15 = lane & 15;
  for (int tsk = wv; tsk < 48; tsk += 8) {           // 6 m-tiles x 8 n-subtiles
    const int m = tsk >> 3, ns = tsk & 7;
    v8f acc = {};
    const float* wrow = &sW[(m * 16 + l15) * 64];
    for (int k = 0; k < 64; k += 4) {
      v2f a, bf;
      a.x = wrow[k + 2 * hf];
      a.y = wrow[k + 2 * hf + 1];
      bf.x = sX[(k + 2 * hf) * 128 + ns * 16 + l15];
      bf.y = sX[(k + 2 * hf + 1) * 128 + ns * 16 + l15];
      acc = __builtin_amdgcn_wmma_f32_16x16x4_f32(
          false, a, false, bf, (short)0, acc, false, false);
    }
    const size_t ncol = pbase + ns * 16 + l15;
    for (int r = 0; r < 8; ++r)
      out[(size_t)(m * 16 + hf * 8 + r) * NPIX + ncol] = acc[r];
  }
}

// =============================================================
// K2: depthwise 3x3 (SAME, zero pad) on qkv; store v; reduce
// the per-head-pair 16x16 gram(q,k) via WMMA and channel
// sum-of-squares of q,k via wave shuffles.
// grid = (256 rows y, 8 zi), block = 256 (one thread per x)
// =============================================================
__global__ __launch_bounds__(256) void k2_dw_reduce(
    const float* __restrict__ qkv, const float* __restrict__ wd1,
    const float* __restrict__ wd2, float* __restrict__ vout,
    float* __restrict__ gramG, float* __restrict__ nqG, float* __restrict__ nkG)
{
  __shared__ float srow[3][264];                 // 3 input rows + x halos
  __shared__ __align__(16) float sQ[16 * QSTR];  // 16 q-ch x 256 px (padded)
  __shared__ __align__(16) float sK[16 * QSTR];  // 16 k-ch x 256 px (padded)
  __shared__ float sacc[1152];                   // 1024 gram + 64 nq + 64 nk
  const int y  = blockIdx.x;
  const int zi = blockIdx.y;
  const int br = zi >> 2;
  const int t  = threadIdx.x;
  const float* base = qkv + (size_t)zi * OCH * NPIX;
  const float* wdw  = br ? wd2 : wd1;

  for (int i = t; i < 1152; i += 256) sacc[i] = 0.f;
  if (t < 3) { srow[t][0] = 0.f; srow[t][257] = 0.f; }   // zero x-pad

  auto dwchan = [&](int c) -> float {
    __syncthreads();   // previous consumers done before refill
    const float* ch = base + (size_t)c * NPIX;
    for (int ky = 0; ky < 3; ++ky) {
      int yy = y + ky - 1;
      float v = 0.f;
      if (yy >= 0 && yy < IMG_H) v = ch[(size_t)yy * IMG_W + t];
      srow[ky][t + 1] = v;
    }
    __syncthreads();
    float w[9];
    for (int i = 0; i < 9; ++i) w[i] = wdw[c * 9 + i];
    float o = 0.f;
    for (int ky = 0; ky < 3; ++ky)
      for (int kx = 0; kx < 3; ++kx)
        o = fmaf(w[ky * 3 + kx], srow[ky][t + kx], o);
    return o;
  };

  const int wv = t >> 5, lane = t & 31;
  const int hf = lane >> 4, l15 = lane & 15;

  for (int hp = 0; hp < 4; ++hp) {               // head-pair = 16 channels
    for (int j = 0; j < 16; ++j) {               // q channels 16*hp ..
      float v = dwchan(hp * 16 + j);
      sQ[j * QSTR + t] = v;
      float sq = v * v;
      for (int off = 16; off > 0; off >>= 1) sq += __shfl_down(sq, off, 32);
      if (lane == 0) atomicAdd(&sacc[1024 + hp * 16 + j], sq);
    }
    for (int j = 0; j < 16; ++j) {               // k channels 64+16*hp ..
      float v = dwchan(64 + hp * 16 + j);
      sK[j * QSTR + t] = v;
      float sk = v * v;
      for (int off = 16; off > 0; off >>= 1) sk += __shfl_down(sk, off, 32);
      if (lane == 0) atomicAdd(&sacc[1088 + hp * 16 + j], sk);
    }
    __syncthreads();
    // gram via WMMA: G(16x16) = Q(16xN) @ K(16xN)^T over this row;
    // wave wv covers the pixel K-chunk [wv*32, wv*32+32)
    v8f acc = {};
    const float* qrow = &sQ[l15 * QSTR];
    const float* krow = &sK[l15 * QSTR];
    for (int s = 0; s < 8; ++s) {
      const int kb = wv * 32 + s * 4 + 2 * hf;   // half-wave K split {0,1|2,3}
      v2f a, bf;
      a.x  = qrow[kb];  a.y  = qrow[kb + 1];     // A: M=q-ch (lane%16)
      bf.x = krow[kb];  bf.y = krow[kb + 1];     // B: N=k-ch (lane%16)
      acc = __builtin_amdgcn_wmma_f32_16x16x4_f32(
          false, a, false, bf, (short)0, acc, false, false);
    }
    // flush the wave's 16x16 partial into the shared gram accumulator
    for (int r = 0; r < 8; ++r)
      atomicAdd(&sacc[hp * 256 + (hf * 8 + r) * 16 + l15], acc[r]);
  }

  // v channels (128..191): depthwise then store for the output GEMM
  float* vdst = vout + (size_t)zi * DCH * NPIX;
  for (int c = 0; c < DCH; ++c) {
    float v = dwchan(128 + c);
    vdst[(size_t)c * NPIX + (size_t)y * IMG_W + t] = v;
  }
  __syncthreads();
  // workgroup partials -> global accumulators
  for (int i = t; i < 1152; i += 256) {
    float v = sacc[i];
    if (i < 1024)      atomicAdd(&gramG[(size_t)zi * 1024 + i], v);
    else if (i < 1088) atomicAdd(&nqG[(size_t)zi * 64 + (i - 1024)], v);
    else               atomicAdd(&nkG[(size_t)zi * 64 + (i - 1088)], v);
  }
}

// =============================================================
// K3: softmax of normalized gram, then fold w_po through the
// block-diagonal attention: M = w_po @ blockdiag(attn).
// grid = 8 (zi), block = 256
// =============================================================
__global__ __launch_bounds__(256) void k3_attn_build(
    const float* __restrict__ gramG, const float* __restrict__ nqG,
    const float* __restrict__ nkG, const float* __restrict__ po1,
    const float* __restrict__ po2, const float* __restrict__ t1,
    const float* __restrict__ t2, float* __restrict__ Mout)
{
  __shared__ float sattn[512];     // [h][cq][ck]
  const int zi = blockIdx.x;
  const int br = zi >> 2;
  const int t  = threadIdx.x;
  if (t < 64) {
    int h = t >> 3, cq = t & 7;
    int hp = h >> 1, hi = h & 1;   // gram stored as [hp][16][16]
    float qn = fmaxf(sqrtf(nqG[zi * 64 + h * 8 + cq]), EPSN);
    float tv = (br ? t2 : t1)[h];
    float s[8];
    float mx = -1e30f;
    for (int ck = 0; ck < 8; ++ck) {
      float kn = fmaxf(sqrtf(nkG[zi * 64 + h * 8 + ck]), EPSN);
      float g  = gramG[zi * 1024 + hp * 256 + (hi * 8 + cq) * 16 + hi * 8 + ck];
      s[ck] = g / (qn * kn) * tv;
      mx = fmaxf(mx, s[ck]);
    }
    float sum = 0.f;
    for (int ck = 0; ck < 8; ++ck) { s[ck] = __expf(s[ck] - mx); sum += s[ck]; }
    float inv = 1.f / sum;
    for (int ck = 0; ck < 8; ++ck) sattn[h * 64 + cq * 8 + ck] = s[ck] * inv;
  }
  __syncthreads();
  const float* po = br ? po2 : po1;
  for (int i = t; i < 4096; i += 256) {
    int o = i >> 6, dch = i & 63;
    int h = dch >> 3, ck = dch & 7;
    float acc = 0.f;
    for (int cq = 0; cq < 8; ++cq)
      acc = fmaf(po[o * 64 + h * 8 + cq], sattn[h * 64 + cq * 8 + ck], acc);
    Mout[(size_t)zi * 4096 + i] = acc;
  }
}

// =============================================================
// K4: out = M @ v_cross (cross-branch V swap), WMMA f32 16x16x4.
// grid = (512 pixel-blocks, 8 zi), block = 256
// =============================================================
__global__ __launch_bounds__(256) void k4_out_gemm(
    const float* __restrict__ Mmat, const float* __restrict__ vbuf,
    float* __restrict__ outp)
{
  __shared__ __align__(16) float sM[64 * 64];    // 16 KB
  __shared__ __align__(16) float sV[64 * 128];   // 32 KB
  const int pb = blockIdx.x;
  const int zi = blockIdx.y;
  const int br = zi >> 2, b = zi & 3;
  const int t  = threadIdx.x;
  const size_t pbase = (size_t)pb * 128;

  const float* Msrc = Mmat + (size_t)zi * 4096;
  const float* vsrc = vbuf + (size_t)((1 - br) * 4 + b) * DCH * NPIX;  // V swap
  float* dst = outp + ((size_t)b * 128 + (size_t)br * 64) * NPIX;

  for (int i = 0; i < 4; ++i)
    ((float4*)sM)[t + i * 256] = ((const float4*)Msrc)[t + i * 256];
  for (int i = 0; i < 8; ++i) {
    int i4 = t + i * 256;
    int ic = i4 >> 5, c4 = i4 & 31;
    ((float4*)sV)[i4] = *(const float4*)(vsrc + (size_t)ic * NPIX + pbase + c4 * 4);
  }
  __syncthreads();

  const int wv = t >> 5, lane = t & 31;
  const int hf = lane >> 4, l15 = lane & 15;
  for (int tsk = wv; tsk < 32; tsk += 8) {          // 4 m-tiles x 8 n-subtiles
    const int m = tsk >> 3, ns = tsk & 7;
    v8f acc = {};
    const float* mrow = &sM[(m * 16 + l15) * 64];
    for (int k = 0; k < 64; k += 4) {
      v2f a, bf;
      a.x = mrow[k + 2 * hf];
      a.y = mrow[k + 2 * hf + 1];
      bf.x = sV[(k + 2 * hf) * 128 + ns * 16 + l15];
      bf.y = sV[(k + 2 * hf + 1) * 128 + ns * 16 + l15];
      acc = __builtin_amdgcn_wmma_f32_16x16x4_f32(
          false, a, false, bf, (short)0, acc, false, false);
    }
    const size_t ncol = pbase + ns * 16 + l15;
    for (int r = 0; r < 8; ++r)
      dst[(size_t)(m * 16 + hf * 8 + r) * NPIX + ncol] = acc[r];
  }
}

// =============================================================
extern "C" void kernel_launch(void* const* d_in, const int* in_sizes, int n_in,
                              void* d_out, int out_size, void* d_ws, size_t ws_size,
                              hipStream_t stream) {
  (void)in_sizes; (void)n_in; (void)out_size; (void)ws_size;
  const float* x   = (const float*)d_in[0];
  const float* wq1 = (const float*)d_in[1];
  const float* wq2 = (const float*)d_in[2];
  const float* wd1 = (const float*)d_in[3];
  const float* wd2 = (const float*)d_in[4];
  const float* po1 = (const float*)d_in[5];
  const float* po2 = (const float*)d_in[6];
  const float* t1  = (const float*)d_in[7];
  const float* t2  = (const float*)d_in[8];
  float* wsf = (float*)d_ws;
  float* out = (float*)d_out;

  float* qkv  = wsf + OFF_QKV;
  float* vbuf = wsf + OFF_V;
  float* gram = wsf + OFF_GRAM;   // gram+nq+nk contiguous (9216 floats)
  float* nq   = wsf + OFF_NQ;
  float* nk   = wsf + OFF_NK;
  float* Mm   = wsf + OFF_M;

  k0_zero<<<36, 256, 0, stream>>>(gram, 9216);
  k1_qkv_gemm<<<dim3(512, 16), 256, 0, stream>>>(x, wq1, wq2, qkv);
  k2_dw_reduce<<<dim3(256, 8), 256, 0, stream>>>(qkv, wd1, wd2, vbuf, gram, nq, nk);
  k3_attn_build<<<8, 256, 0, stream>>>(gram, nq, nk, po1, po2, t1, t2, Mm);
  k4_out_gemm<<<dim3(512, 8), 256, 0, stream>>>(Mm, vbuf, out);
}